// AttentionCropLayer_33870112096711
// MI455X (gfx1250) — compile-verified
//
#include <hip/hip_runtime.h>
#include <math.h>

// AttentionCropLayer: soft-mask crop + align-corners bilinear resize.
//   images: [B, 3, 448, 448] f32, locs: [B, 3] f32 -> out: [B, 3, 224, 224] f32
//
// Bandwidth-bound (~190 MB moved, ~0.3 GFLOP). Strategy:
//  - one 224-thread block per (b, out_row); the 6 source rows (2 rows x 3 ch)
//    are staged into LDS with gfx1250 GLOBAL_LOAD_ASYNC_TO_LDS_B64 (each of
//    224 lanes moves 8B -> 1792B row per instruction), completed with
//    s_wait_asynccnt + barrier.
//  - bilinear taps become LDS gathers (near stride-1 -> conflict free).
//  - sigmoid masks folded into the 4 tap weights (no 448x448 mask material-
//    ization, no second pass over the image).
//  - output streamed with nontemporal stores so the 154MB input (fits in the
//    192MB L2) keeps the cache.

#define IN_SZ  448
#define OUT_SZ 224
#define NCH    3

__device__ __forceinline__ float sig10(float v) {
    // sigmoid(10*v)
    return 1.0f / (1.0f + expf(-10.0f * v));
}

__global__ __launch_bounds__(OUT_SZ)
void crop_resize_kernel(const float* __restrict__ images,
                        const float* __restrict__ locs,
                        float* __restrict__ out)
{
    __shared__ float rows[NCH][2][IN_SZ];   // 10752 B LDS

    const int out_r = blockIdx.x;   // 0..223
    const int b     = blockIdx.y;   // sample
    const int tid   = threadIdx.x;  // 0..223  (output column)

    // ---- per-sample crop parameters (uniform across block) ----
    float tx = locs[b * 3 + 0];
    float ty = locs[b * 3 + 1];
    float tl = locs[b * 3 + 2];
    const float third = (float)IN_SZ / 3.0f;
    tl = fmaxf(tl, third);
    tx = fminf(fmaxf(tx, tl), (float)IN_SZ - tl);   // jnp.clip = min(max(x,lo),hi)
    ty = fminf(fmaxf(ty, tl), (float)IN_SZ - tl);
    const float w_off = fminf(fmaxf(floorf(tx - tl), 0.0f), (float)IN_SZ);
    const float h_off = fminf(fmaxf(floorf(ty - tl), 0.0f), (float)IN_SZ);
    const float w_end = fminf(fmaxf(floorf(tx + tl), 0.0f), (float)IN_SZ);
    const float h_end = fminf(fmaxf(floorf(ty + tl), 0.0f), (float)IN_SZ);

    // ---- row (first spatial axis) sample coordinate for this block ----
    const float t_r = (float)out_r * (1.0f / (float)(OUT_SZ - 1));
    const float r   = w_off + t_r * (w_end - 1.0f - w_off);
    const int   r0  = (int)floorf(r);                 // >= 0 by construction
    const int   r1  = min(r0 + 1, IN_SZ - 1);
    const float fr  = r - (float)r0;

    // ---- async-stage the 6 needed source rows into LDS ----
    {
        const unsigned laneByte = (unsigned)tid * 8u;       // 224 lanes x 8B = 1792B
        const size_t imgBase = (size_t)b * NCH * IN_SZ * IN_SZ;
#pragma unroll
        for (int ch = 0; ch < NCH; ++ch) {
            const float* g0 = images + imgBase + ((size_t)ch * IN_SZ + (size_t)r0) * IN_SZ;
            const float* g1 = images + imgBase + ((size_t)ch * IN_SZ + (size_t)r1) * IN_SZ;
            unsigned l0 = (unsigned)(size_t)(&rows[ch][0][0]) + laneByte; // LDS byte addr
            unsigned l1 = (unsigned)(size_t)(&rows[ch][1][0]) + laneByte;
            unsigned long long a0 = (unsigned long long)(size_t)g0 + laneByte;
            unsigned long long a1 = (unsigned long long)(size_t)g1 + laneByte;
            asm volatile("global_load_async_to_lds_b64 %0, %1, off"
                         :: "v"(l0), "v"(a0) : "memory");
            asm volatile("global_load_async_to_lds_b64 %0, %1, off"
                         :: "v"(l1), "v"(a1) : "memory");
        }
        asm volatile("s_wait_asynccnt 0" ::: "memory");  // this wave's copies done
    }
    __syncthreads();                                     // all 7 waves' copies visible

    // ---- column (second spatial axis) sample coordinate per thread ----
    const float t_c = (float)tid * (1.0f / (float)(OUT_SZ - 1));
    const float c   = h_off + t_c * (h_end - 1.0f - h_off);
    const int   c0  = (int)floorf(c);
    const int   c1  = min(c0 + 1, IN_SZ - 1);
    const float fc  = c - (float)c0;

    // ---- sigmoid mask folded into the 4 bilinear tap weights ----
    const float mr0 = sig10((float)r0 - w_off) - sig10((float)r0 - w_end);
    const float mr1 = sig10((float)r1 - w_off) - sig10((float)r1 - w_end);
    const float mc0 = sig10((float)c0 - h_off) - sig10((float)c0 - h_end);
    const float mc1 = sig10((float)c1 - h_off) - sig10((float)c1 - h_end);

    const float w00 = mr0 * mc0 * (1.0f - fr) * (1.0f - fc);
    const float w01 = mr0 * mc1 * (1.0f - fr) * fc;
    const float w10 = mr1 * mc0 * fr * (1.0f - fc);
    const float w11 = mr1 * mc1 * fr * fc;

    // ---- gather from LDS, stream result (write-once -> nontemporal) ----
#pragma unroll
    for (int ch = 0; ch < NCH; ++ch) {
        const float v00 = rows[ch][0][c0];
        const float v01 = rows[ch][0][c1];
        const float v10 = rows[ch][1][c0];
        const float v11 = rows[ch][1][c1];
        const float res = v00 * w00 + v01 * w01 + v10 * w10 + v11 * w11;
        float* dst = out + (((size_t)b * NCH + ch) * OUT_SZ + (size_t)out_r) * OUT_SZ + tid;
        __builtin_nontemporal_store(res, dst);
    }
}

extern "C" void kernel_launch(void* const* d_in, const int* in_sizes, int n_in,
                              void* d_out, int out_size, void* d_ws, size_t ws_size,
                              hipStream_t stream) {
    (void)n_in; (void)out_size; (void)d_ws; (void)ws_size;
    const float* images = (const float*)d_in[0];
    const float* locs   = (const float*)d_in[1];
    float*       out    = (float*)d_out;

    const int B = in_sizes[1] / 3;           // locs is [B,3] -> B = 64
    dim3 grid(OUT_SZ, B);                    // (out_row, sample)
    crop_resize_kernel<<<grid, OUT_SZ, 0, stream>>>(images, locs, out);
}